// GINConvLayer_21801253995167
// MI455X (gfx1250) — compile-verified
//
#include <hip/hip_runtime.h>

#define N_NODES 10000
#define N_EDGES 320000
#define DIM     256
#define BN_EPS  1e-5f

typedef float v2f __attribute__((ext_vector_type(2)));
typedef float v8f __attribute__((ext_vector_type(8)));

// ---------------------------------------------------------------------------
// Kernel 1: h = (1 + eps) * x      (vectorized float4, fully coalesced)
// ---------------------------------------------------------------------------
__global__ void init_h_kernel(const float* __restrict__ x,
                              const float* __restrict__ eps,
                              float* __restrict__ h) {
    int i = blockIdx.x * blockDim.x + threadIdx.x;
    int total = (N_NODES * DIM) / 4;
    if (i >= total) return;
    float e = 1.0f + eps[0];
    float4 v = ((const float4*)x)[i];
    v.x *= e; v.y *= e; v.z *= e; v.w *= e;
    ((float4*)h)[i] = v;
}

// ---------------------------------------------------------------------------
// Kernel 2: scatter-add  h[row] += x[col] * edge_embed[e]
// 4 edges per 256-thread block; 64 lanes x float4 = 256 features per edge.
// edge_embed streamed once (327 MB); x gathers hit L2; atomics stay in L2.
// ---------------------------------------------------------------------------
__global__ void edge_scatter_kernel(const float* __restrict__ x,
                                    const int*   __restrict__ row,
                                    const int*   __restrict__ col,
                                    const float* __restrict__ ee,
                                    float* __restrict__ h) {
    int t      = threadIdx.x;
    int eLocal = t >> 6;            // 0..3
    int f      = (t & 63) << 2;     // feature offset 0..252
    int e      = blockIdx.x * 4 + eLocal;
    if (e >= N_EDGES) return;
    int r = row[e];
    int c = col[e];
    float4 xv = *(const float4*)(x  + (size_t)c * DIM + f);
    float4 ev = *(const float4*)(ee + (size_t)e * DIM + f);
    float* dst = h + (size_t)r * DIM + f;
    atomicAdd(dst + 0, xv.x * ev.x);
    atomicAdd(dst + 1, xv.y * ev.y);
    atomicAdd(dst + 2, xv.z * ev.z);
    atomicAdd(dst + 3, xv.w * ev.w);
}

// ---------------------------------------------------------------------------
// WMMA GEMM: C[MxDIM] = op(A)[MxDIM] @ W[DIMxDIM] + bias
// One 16x16 output tile per wave (8 waves / 256-thread block).
// FUSE_BN_RELU: A element -> relu(A*scale[k] + shift[k])  (BatchNorm+ReLU).
// Uses V_WMMA_F32_16X16X4_F32, full f32 precision (matches f32 reference).
//
// A 16x4 f32 layout: lanes 0-15 hold M=lane (K=0,1), lanes 16-31 M=lane-16
// (K=2,3) -> per lane load A[m][k+koff], A[m][k+koff+1], koff = (lane>=16)*2.
// C/D 16x16 f32 layout: VGPR r -> row tileM + r + (lane>=16)*8, col = lane&15.
// ---------------------------------------------------------------------------
template <bool FUSE_BN_RELU>
__global__ void gemm_wmma_kernel(const float* __restrict__ A,
                                 const float* __restrict__ W,
                                 const float* __restrict__ bias,
                                 const float* __restrict__ scale,
                                 const float* __restrict__ shift,
                                 float* __restrict__ C) {
    int wave = threadIdx.x >> 5;
    int lane = threadIdx.x & 31;
    int gw   = blockIdx.x * 8 + wave;        // 0..9999 tiles (625 x 16)
    int tileM = (gw >> 4) << 4;              // 0..9984 step 16
    int tileN = (gw & 15) << 4;              // 0..240  step 16
    int lm    = lane & 15;
    int koff  = (lane >> 4) << 1;            // 0 or 2

    const float* arow = A + (size_t)(tileM + lm) * DIM;

    v8f acc = {0.f, 0.f, 0.f, 0.f, 0.f, 0.f, 0.f, 0.f};

#pragma unroll 8
    for (int k = 0; k < DIM; k += 4) {
        int ka = k + koff;
        v2f a;
        a.x = arow[ka];
        a.y = arow[ka + 1];
        if (FUSE_BN_RELU) {
            a.x = fmaxf(fmaf(a.x, scale[ka],     shift[ka]),     0.0f);
            a.y = fmaxf(fmaf(a.y, scale[ka + 1], shift[ka + 1]), 0.0f);
        }
        v2f b;
        b.x = W[(size_t)ka       * DIM + tileN + lm];
        b.y = W[(size_t)(ka + 1) * DIM + tileN + lm];
        acc = __builtin_amdgcn_wmma_f32_16x16x4_f32(
            /*neg_a=*/false, a, /*neg_b=*/false, b,
            /*c_mod=*/(short)0, acc, /*reuse_a=*/false, /*reuse_b=*/false);
    }

    int rowBase = tileM + ((lane >> 4) << 3);
    float bv = bias[tileN + lm];
#pragma unroll
    for (int r = 0; r < 8; ++r) {
        C[(size_t)(rowBase + r) * DIM + tileN + lm] = acc[r] + bv;
    }
}

// ---------------------------------------------------------------------------
// Per-feature sum / sum-of-squares over N rows (coalesced row-major reads,
// register accumulation, one atomic pair per block per feature).
// ---------------------------------------------------------------------------
__global__ void col_stats_kernel(const float* __restrict__ h1,
                                 float* __restrict__ sum,
                                 float* __restrict__ sumsq) {
    int d = threadIdx.x;                      // 256 features
    float s = 0.0f, s2 = 0.0f;
    for (int i = blockIdx.x; i < N_NODES; i += gridDim.x) {
        float v = h1[(size_t)i * DIM + d];
        s  += v;
        s2 += v * v;
    }
    atomicAdd(&sum[d],   s);
    atomicAdd(&sumsq[d], s2);
}

__global__ void bn_finalize_kernel(const float* __restrict__ sum,
                                   const float* __restrict__ sumsq,
                                   const float* __restrict__ gamma,
                                   const float* __restrict__ beta,
                                   float* __restrict__ scale,
                                   float* __restrict__ shift) {
    int d = threadIdx.x;
    float invN = 1.0f / (float)N_NODES;
    float mean = sum[d] * invN;
    float var  = sumsq[d] * invN - mean * mean;
    float s    = gamma[d] * rsqrtf(var + BN_EPS);
    scale[d] = s;
    shift[d] = beta[d] - mean * s;
}

// ---------------------------------------------------------------------------
// Launch: memset stats -> init_h -> edge_scatter -> GEMM1 -> stats ->
//         bn_finalize -> GEMM2(fused BN+ReLU) -> d_out
// ---------------------------------------------------------------------------
extern "C" void kernel_launch(void* const* d_in, const int* in_sizes, int n_in,
                              void* d_out, int out_size, void* d_ws, size_t ws_size,
                              hipStream_t stream) {
    const float* x     = (const float*)d_in[0];
    const int*   eidx  = (const int*)  d_in[1];   // [2, E]
    const float* ee    = (const float*)d_in[2];
    const float* W1    = (const float*)d_in[3];
    const float* b1    = (const float*)d_in[4];
    const float* gamma = (const float*)d_in[5];
    const float* beta  = (const float*)d_in[6];
    const float* W2    = (const float*)d_in[7];
    const float* b2    = (const float*)d_in[8];
    const float* eps   = (const float*)d_in[9];

    const int* row = eidx;
    const int* col = eidx + N_EDGES;

    // Workspace layout
    float* h     = (float*)d_ws;                         // N*D
    float* h1    = h  + (size_t)N_NODES * DIM;           // N*D
    float* stats = h1 + (size_t)N_NODES * DIM;           // sum(256), sumsq(256)
    float* sum   = stats;
    float* sumsq = stats + DIM;
    float* scale = stats + 2 * DIM;                      // 256
    float* shift = stats + 3 * DIM;                      // 256

    hipMemsetAsync(stats, 0, 2 * DIM * sizeof(float), stream);

    int initBlocks = (N_NODES * DIM / 4 + 255) / 256;    // 2500
    init_h_kernel<<<initBlocks, 256, 0, stream>>>(x, eps, h);

    edge_scatter_kernel<<<N_EDGES / 4, 256, 0, stream>>>(x, row, col, ee, h);

    int gemmBlocks = (N_NODES / 16) * (DIM / 16) / 8;    // 1250
    gemm_wmma_kernel<false><<<gemmBlocks, 256, 0, stream>>>(
        h, W1, b1, nullptr, nullptr, h1);

    col_stats_kernel<<<256, 256, 0, stream>>>(h1, sum, sumsq);
    bn_finalize_kernel<<<1, 256, 0, stream>>>(sum, sumsq, gamma, beta, scale, shift);

    gemm_wmma_kernel<true><<<gemmBlocks, 256, 0, stream>>>(
        h1, W2, b2, scale, shift, (float*)d_out);
}